// HeteroLabelEmbeddingGNN_19662360281239
// MI455X (gfx1250) — compile-verified
//
#include <hip/hip_runtime.h>
#include <math.h>

// ---------------- problem constants (match reference) ----------------
#define NN     50000
#define HID    64
#define HEADS  4
#define CH     16
#define RREL   4
#define EDGES  200000
#define ENE    (EDGES + NN)     // edges + self loops = 250000
#define SLOPE  0.2f
#define LN_EPS 1e-5f

typedef __attribute__((ext_vector_type(2))) float v2f;
typedef __attribute__((ext_vector_type(8))) float v8f;
typedef __attribute__((ext_vector_type(4))) int   i32x4;

typedef i32x4 __attribute__((address_space(1)))* gasp;   // global side (b128 chunks)
typedef i32x4 __attribute__((address_space(3)))* lasp;   // LDS side   (b128 chunks)

__device__ __forceinline__ float gelu_exact(float x) {
  return 0.5f * x * (1.0f + erff(x * 0.70710678118654752440f));
}

// float atomic max via sign-split int/uint ordering (init value must be -inf)
__device__ __forceinline__ void atomic_max_f32(float* addr, float val) {
  if (val >= 0.0f) {
    atomicMax((int*)addr, __float_as_int(val));
  } else {
    atomicMin((unsigned int*)addr, __float_as_uint(val));
  }
}

__device__ __forceinline__ float wave_sum32(float v) {
#pragma unroll
  for (int m = 16; m >= 1; m >>= 1) v += __shfl_xor(v, m, 32);
  return v;
}

// ================= GEMM: Y[N,64] = X[N,64] @ W[64,64] (+bias) =================
// One 16-row tile per wave; both W (shared) and the per-wave X tile are staged
// into LDS with GLOBAL_LOAD_ASYNC_TO_LDS_B128 (ASYNCcnt), then consumed via
// ds_load for the WMMA fragments. 64 f32 WMMAs per wave.
//   W stride 72 floats (288B): 16B-aligned rows, B-frag lane-halves hit
//     disjoint 16-bank groups.
//   X stride 68 floats (272B): 16B-aligned rows, A-frag b64 reads are fully
//     bank-conflict-free (bank = 4*row + kbase mod 64, distinct for all lanes).
#define NTILES      (NN / 16)                       // 3125 (exact)
#define GEMM_WAVES  4
#define GEMM_BLOCKS ((NTILES + GEMM_WAVES - 1) / GEMM_WAVES)
#define WSTRIDE     72
#define XSTRIDE     68

__global__ __launch_bounds__(128) void k_gemm64(const float* __restrict__ X,
                                                const float* __restrict__ W,
                                                const float* __restrict__ bias,
                                                float* __restrict__ Y) {
  __shared__ float wlds[64 * WSTRIDE];                 // 18432 B
  __shared__ float xtile[GEMM_WAVES * 16 * XSTRIDE];   // 17408 B
  const int tid  = threadIdx.x;
  const int wave = tid >> 5;
  const int lane = tid & 31;
  const int tile = blockIdx.x * GEMM_WAVES + wave;     // wave-uniform

  // ---- async DMA: W (64x64 f32 = 1024 x 16B chunks, all 128 threads) ----
#pragma unroll
  for (int it = 0; it < 8; ++it) {
    const int chunk = tid + it * 128;                  // [0,1024)
    const int row = chunk >> 4, cc = chunk & 15;
    __builtin_amdgcn_global_load_async_to_lds_b128(
        (gasp)(W + row * 64 + cc * 4),
        (lasp)(wlds + row * WSTRIDE + cc * 4), 0, 0);
  }
  // ---- async DMA: this wave's 16x64 X tile (256 x 16B chunks, 32 lanes) ----
  float* xt = xtile + wave * 16 * XSTRIDE;
  if (tile < NTILES) {
    const float* xbase = X + (size_t)tile * 16 * HID;
#pragma unroll
    for (int it = 0; it < 8; ++it) {
      const int chunk = lane + it * 32;                // [0,256)
      const int row = chunk >> 4, cc = chunk & 15;
      __builtin_amdgcn_global_load_async_to_lds_b128(
          (gasp)(xbase + row * 64 + cc * 4),
          (lasp)(xt + row * XSTRIDE + cc * 4), 0, 0);
    }
  }
  asm volatile("s_wait_asynccnt 0x0" ::: "memory");    // my wave's DMAs done
  __syncthreads();                                     // W visible to all waves
  if (tile >= NTILES) return;                          // whole-wave exit: EXEC stays all-ones

  const int lh    = lane & 15;            // M (for A) / N (for B,C)
  const int khalf = (lane >> 4) * 2;      // K sub-offset: 0 or 2

  // A fragments from LDS: 16 K-chunks of 16x4
  // (lane<16: row lh K=4kk,4kk+1; lane>=16: row lh K=4kk+2,4kk+3)
  const float* xt_l = xt + lh * XSTRIDE + khalf;
  v2f a[16];
#pragma unroll
  for (int kk = 0; kk < 16; ++kk)
    a[kk] = *(const v2f*)(xt_l + kk * 4);

#pragma unroll
  for (int j = 0; j < 4; ++j) {           // four 16-wide N tiles
    v8f c = {0.f, 0.f, 0.f, 0.f, 0.f, 0.f, 0.f, 0.f};
#pragma unroll
    for (int kk = 0; kk < 16; ++kk) {
      const int kb = kk * 4 + khalf;
      v2f b;
      b.x = wlds[(kb + 0) * WSTRIDE + j * 16 + lh];
      b.y = wlds[(kb + 1) * WSTRIDE + j * 16 + lh];
      c = __builtin_amdgcn_wmma_f32_16x16x4_f32(false, a[kk], false, b,
                                                (short)0, c, false, false);
    }
    const int col = j * 16 + lh;
    if (bias != nullptr) {
      const float bv = bias[col];
#pragma unroll
      for (int r2 = 0; r2 < 8; ++r2) c[r2] += bv;
    }
    const int rowbase = tile * 16 + (lane >> 4) * 8;  // C/D: M split across lane halves
#pragma unroll
    for (int r2 = 0; r2 < 8; ++r2)
      Y[(size_t)(rowbase + r2) * HID + col] = c[r2];
  }
}

// ================= entry: out = gelu(LN(Yin)) — one wave per row =================
__global__ void k_ln_entry(const float* __restrict__ Yin,
                           const float* __restrict__ g,
                           const float* __restrict__ b,
                           float* __restrict__ out) {
  const int wave = (int)((blockIdx.x * blockDim.x + threadIdx.x) >> 5);
  const int lane = threadIdx.x & 31;
  if (wave >= NN) return;
  const float* row = Yin + (size_t)wave * HID;
  const float v0 = row[lane], v1 = row[lane + 32];
  const float mu = wave_sum32(v0 + v1) * (1.0f / 64.0f);
  const float d0 = v0 - mu, d1 = v1 - mu;
  const float var = wave_sum32(d0 * d0 + d1 * d1) * (1.0f / 64.0f);
  const float rstd = rsqrtf(var + LN_EPS);
  float* orow = out + (size_t)wave * HID;
  orow[lane]      = gelu_exact(d0 * rstd * g[lane]      + b[lane]);
  orow[lane + 32] = gelu_exact(d1 * rstd * g[lane + 32] + b[lane + 32]);
}

// ====== post-layer: out = LN(gelu((acc + sum_r bias_r)/R)) — one wave per row ======
__global__ void k_gelu_ln(const float* __restrict__ acc,
                          const float* __restrict__ biasR,   // [R][64]
                          const float* __restrict__ g,
                          const float* __restrict__ b,
                          float* __restrict__ out) {
  const int wave = (int)((blockIdx.x * blockDim.x + threadIdx.x) >> 5);
  const int lane = threadIdx.x & 31;
  if (wave >= NN) return;
  const int c0 = lane, c1 = lane + 32;
  const float bs0 = biasR[c0] + biasR[64 + c0] + biasR[128 + c0] + biasR[192 + c0];
  const float bs1 = biasR[c1] + biasR[64 + c1] + biasR[128 + c1] + biasR[192 + c1];
  const float* row = acc + (size_t)wave * HID;
  const float u0 = gelu_exact((row[c0] + bs0) * 0.25f);
  const float u1 = gelu_exact((row[c1] + bs1) * 0.25f);
  const float mu = wave_sum32(u0 + u1) * (1.0f / 64.0f);
  const float d0 = u0 - mu, d1 = u1 - mu;
  const float var = wave_sum32(d0 * d0 + d1 * d1) * (1.0f / 64.0f);
  const float rstd = rsqrtf(var + LN_EPS);
  float* orow = out + (size_t)wave * HID;
  orow[c0] = d0 * rstd * g[c0] + b[c0];
  orow[c1] = d1 * rstd * g[c1] + b[c1];
}

// ================= init kernels (ws is poisoned; must self-initialize) =================
__global__ void k_fill(float* __restrict__ p, float v, int n) {
  const int t = blockIdx.x * blockDim.x + threadIdx.x;
  if (t < n) p[t] = v;
}
__global__ void k_init_softmax(float* __restrict__ emax, float* __restrict__ den, int n) {
  const int t = blockIdx.x * blockDim.x + threadIdx.x;
  if (t < n) { emax[t] = -__builtin_inff(); den[t] = 0.0f; }
}

// ================= edge pass 1: logits e[i,h] + segment max over dst =================
__global__ void k_edge_logits(const int* __restrict__ src, const int* __restrict__ dst,
                              const float* __restrict__ xl, const float* __restrict__ xr,
                              const float* __restrict__ att,   // [HEADS*CH] for (layer, r)
                              float* __restrict__ ebuf, float* __restrict__ emax) {
  const int t = blockIdx.x * blockDim.x + threadIdx.x;
  if (t >= ENE * HEADS) return;
  const int i = t >> 2, h = t & 3;
  const int s = (i < EDGES) ? src[i] : (i - EDGES);   // self loops appended
  const int d = (i < EDGES) ? dst[i] : (i - EDGES);
  const float* pl = xl + (size_t)s * HID + h * CH;
  const float* pr = xr + (size_t)d * HID + h * CH;
  const float* pa = att + h * CH;
  float accv = 0.0f;
#pragma unroll
  for (int c = 0; c < CH; ++c) {
    float m = pl[c] + pr[c];
    m = (m > 0.0f) ? m : SLOPE * m;        // GATv2 leaky relu
    accv += m * pa[c];
  }
  ebuf[t] = accv;
  atomic_max_f32(emax + d * HEADS + h, accv);
}

// ================= edge pass 2: a = exp(e - emax[dst]); segment sum =================
__global__ void k_edge_exp(const int* __restrict__ dst,
                           float* __restrict__ ebuf,
                           const float* __restrict__ emax, float* __restrict__ den) {
  const int t = blockIdx.x * blockDim.x + threadIdx.x;
  if (t >= ENE * HEADS) return;
  const int i = t >> 2, h = t & 3;
  const int d = (i < EDGES) ? dst[i] : (i - EDGES);
  const float a = expf(ebuf[t] - emax[d * HEADS + h]);
  ebuf[t] = a;
  atomicAdd(den + d * HEADS + h, a);
}

// ================= edge pass 3: scatter alpha * xl[src] into accumulator =================
__global__ void k_edge_scatter(const int* __restrict__ src, const int* __restrict__ dst,
                               const float* __restrict__ ebuf, const float* __restrict__ den,
                               const float* __restrict__ xl, float* __restrict__ acc) {
  const int t = blockIdx.x * blockDim.x + threadIdx.x;
  if (t >= ENE * HEADS) return;
  const int i = t >> 2, h = t & 3;
  const int s = (i < EDGES) ? src[i] : (i - EDGES);
  const int d = (i < EDGES) ? dst[i] : (i - EDGES);
  const float alpha = ebuf[t] / den[d * HEADS + h];
  const float* pl = xl + (size_t)s * HID + h * CH;
  float* pd = acc + (size_t)d * HID + h * CH;
#pragma unroll
  for (int c = 0; c < CH; ++c)
    atomicAdd(pd + c, pl[c] * alpha);
}

// =============================== host-side launch ===============================
extern "C" void kernel_launch(void* const* d_in, const int* in_sizes, int n_in,
                              void* d_out, int out_size, void* d_ws, size_t ws_size,
                              hipStream_t stream) {
  const float* emb        = (const float*)d_in[0];
  const float* entry_w    = (const float*)d_in[1];
  const float* entry_b    = (const float*)d_in[2];
  const float* entry_ln_g = (const float*)d_in[3];
  const float* entry_ln_b = (const float*)d_in[4];
  const float* conv_wl    = (const float*)d_in[5];   // [2,4,64,64]
  const float* conv_wr    = (const float*)d_in[6];   // [2,4,64,64]
  const float* conv_att   = (const float*)d_in[7];   // [2,4,4,16]
  const float* conv_bias  = (const float*)d_in[8];   // [2,4,64]
  const float* norm_g     = (const float*)d_in[9];
  const float* norm_b     = (const float*)d_in[10];
  const int*   ei         = (const int*)d_in[11];    // [4,2,200000]

  // workspace layout (all f32): x | xl | xr | acc | ebuf | emax | den   (~57 MB)
  float* ws   = (float*)d_ws;
  float* x    = ws;
  float* xl   = x    + (size_t)NN * HID;
  float* xr   = xl   + (size_t)NN * HID;
  float* acc  = xr   + (size_t)NN * HID;
  float* ebuf = acc  + (size_t)NN * HID;
  float* emax = ebuf + (size_t)ENE * HEADS;
  float* den  = emax + (size_t)NN * HEADS;

  const int LN_BLOCKS   = NN / 8;                       // 8 waves (rows) per 256-thread block
  const int EH          = ENE * HEADS;                  // 1,000,000
  const int EDGE_BLOCKS = (EH + 255) / 256;
  const int NH_BLOCKS   = (NN * HEADS + 255) / 256;
  const int ND_BLOCKS   = (NN * HID + 255) / 256;

  // ---- entry: x = gelu(LN(emb @ entry_w + entry_b)) ----
  k_gemm64<<<GEMM_BLOCKS, 128, 0, stream>>>(emb, entry_w, entry_b, xl);
  k_ln_entry<<<LN_BLOCKS, 256, 0, stream>>>(xl, entry_ln_g, entry_ln_b, x);

  for (int layer = 0; layer < 2; ++layer) {
    k_fill<<<ND_BLOCKS, 256, 0, stream>>>(acc, 0.0f, NN * HID);
    for (int r = 0; r < RREL; ++r) {
      const float* wl   = conv_wl  + (size_t)(layer * RREL + r) * HID * HID;
      const float* wr_  = conv_wr  + (size_t)(layer * RREL + r) * HID * HID;
      const float* att  = conv_att + (size_t)(layer * RREL + r) * HEADS * CH;
      const int*   srcp = ei + (size_t)r * 2 * EDGES;
      const int*   dstp = srcp + EDGES;

      k_gemm64<<<GEMM_BLOCKS, 128, 0, stream>>>(x, wl,  nullptr, xl);
      k_gemm64<<<GEMM_BLOCKS, 128, 0, stream>>>(x, wr_, nullptr, xr);
      k_init_softmax<<<NH_BLOCKS, 256, 0, stream>>>(emax, den, NN * HEADS);
      k_edge_logits <<<EDGE_BLOCKS, 256, 0, stream>>>(srcp, dstp, xl, xr, att, ebuf, emax);
      k_edge_exp    <<<EDGE_BLOCKS, 256, 0, stream>>>(dstp, ebuf, emax, den);
      k_edge_scatter<<<EDGE_BLOCKS, 256, 0, stream>>>(srcp, dstp, ebuf, den, xl, acc);
    }
    const float* bb   = conv_bias + (size_t)layer * RREL * HID;
    float*       dest = (layer == 1) ? (float*)d_out : x;
    k_gelu_ln<<<LN_BLOCKS, 256, 0, stream>>>(acc, bb, norm_g, norm_b, dest);
  }
}